// yolo_layer_6854767805041
// MI455X (gfx1250) — compile-verified
//
#include <hip/hip_runtime.h>
#include <math.h>

#define Bd 16
#define Ad 3
#define CLS 80
#define Gd 52
#define NT 256
#define CH (CLS + 5)                  // 85
#define NCELL (Bd * Ad * Gd * Gd)     // 129792 = 507 * 256 exactly
#define EPSF 1e-16f
#define IGNORE_THRESH 0.5f

typedef __attribute__((ext_vector_type(2))) float v2f;
typedef __attribute__((ext_vector_type(8))) float v8f;

__device__ __forceinline__ float fast_sigmoid(float v) {
    // 1/(1+e^-v) with v_exp_f32 + v_rcp_f32 (transcendental units, ~2 ULP)
    return __builtin_amdgcn_rcpf(1.0f + __expf(-v));
}

// ---------------------------------------------------------------- init
__global__ void yolo_init_kernel(int* __restrict__ map, float* __restrict__ out) {
    int idx = blockIdx.x * blockDim.x + threadIdx.x;
    if (idx < NCELL) map[idx] = -1;
    if (idx == 0) out[0] = 0.0f;
}

// ------------------------------------------------------- target assignment
// One thread per target. Valid targets (tn in [0,2]) claim their cell;
// duplicates resolved by atomicMax (JAX scatter-set w/ dup indices is
// "one winner" — we deterministically pick the largest target index).
__global__ void yolo_assign_kernel(const float* __restrict__ target,
                                   int* __restrict__ map) {
    int t = threadIdx.x;              // blockDim.x == NT
    const float* tg = target + t * 7;
    float tn = tg[6];
    if (tn >= 0.0f && tn <= 2.0f) {
        int b = (int)tg[0];
        int n = (int)tn;
        n = n < 0 ? 0 : (n > Ad - 1 ? Ad - 1 : n);
        int gi = (int)(tg[2] * (float)Gd);
        int gj = (int)(tg[3] * (float)Gd);
        int cell = ((b * Ad + n) * Gd + gj) * Gd + gi;
        atomicMax(&map[cell], t);
    }
}

// ---------------------------------------------------------------- main
__global__ void __launch_bounds__(256)
yolo_grad_sq_kernel(const float* __restrict__ x,
                    const float* __restrict__ anchors,
                    const float* __restrict__ target,
                    const int*   __restrict__ map,
                    float*       __restrict__ out) {
    // Target boxes cached in LDS (corner form + area): 6 KB
    __shared__ float s_tb[NT], s_x1[NT], s_y1[NT], s_x2[NT], s_y2[NT], s_ar[NT];
    {
        int lt = threadIdx.x;         // blockDim.x == NT == 256
        const float* tg = target + lt * 7;
        float tb = tg[0], tx = tg[2], ty = tg[3], tw = tg[4], th = tg[5];
        s_tb[lt] = tb;
        s_x1[lt] = tx - tw * 0.5f;
        s_y1[lt] = ty - th * 0.5f;
        s_x2[lt] = tx + tw * 0.5f;
        s_y2[lt] = ty + th * 0.5f;
        s_ar[lt] = tw * th;
    }
    __syncthreads();

    int idx = blockIdx.x * 256 + threadIdx.x;      // always < NCELL (507*256)
    int i = idx % Gd;
    int j = (idx / Gd) % Gd;
    int a = (idx / (Gd * Gd)) % Ad;
    int b = idx / (Gd * Gd * Ad);

    const float invG = 1.0f / (float)Gd;
    const size_t cs = (size_t)(Gd * Gd);
    size_t base = (size_t)(b * (Ad * CH) + a * CH) * cs + (size_t)j * Gd + (size_t)i;

    float rdx   = x[base + 0 * cs];
    float rdy   = x[base + 1 * cs];
    float pdw   = x[base + 2 * cs];
    float pdh   = x[base + 3 * cs];
    float rconf = x[base + 4 * cs];

    float pdx  = fast_sigmoid(rdx);
    float pdy  = fast_sigmoid(rdy);
    float conf = fast_sigmoid(rconf);

    float aw = anchors[a * 2 + 0];
    float ah = anchors[a * 2 + 1];

    float px = (pdx + (float)i) * invG;
    float py = (pdy + (float)j) * invG;
    float pw = __expf(pdw) * aw * invG;
    float ph = __expf(pdh) * ah * invG;

    float p_x1 = px - pw * 0.5f, p_y1 = py - ph * 0.5f;
    float p_x2 = px + pw * 0.5f, p_y2 = py + ph * 0.5f;
    float parea = pw * ph;
    float bf = (float)b;

    // Ignore-mask: max IOU over targets of this batch (segment_max + clip0).
    // Hot loop (~33M iters total): v_rcp_f32 instead of precise divide.
    float maxiou = 0.0f;
#pragma unroll 4
    for (int t = 0; t < NT; ++t) {
        float iw = fminf(p_x2, s_x2[t]) - fmaxf(p_x1, s_x1[t]);
        float ih = fminf(p_y2, s_y2[t]) - fmaxf(p_y1, s_y1[t]);
        iw = fmaxf(iw, 0.0f);
        ih = fmaxf(ih, 0.0f);
        float inter = iw * ih;
        float iou = inter * __builtin_amdgcn_rcpf(parea + s_ar[t] - inter + EPSF);
        iou = (s_tb[t] == bf) ? iou : 0.0f;
        maxiou = fmaxf(maxiou, iou);
    }

    float contrib;
    int win = map[idx];
    if (win >= 0) {
        // Target cell: x/y/w/h/conf + 80 class grads
        const float* tg = target + win * 7;
        float tx = tg[2], ty = tg[3], tw = tg[4], th = tg[5];
        int label = (int)tg[1];
        float scale = 2.0f - tw * th;
        float gx = tx * (float)Gd, gy = ty * (float)Gd;
        float gxv = scale * (pdx - (gx - floorf(gx)));
        float gyv = scale * (pdy - (gy - floorf(gy)));
        float gwv = scale * (pdw - __logf(tw * (float)Gd / aw + EPSF));
        float ghv = scale * (pdh - __logf(th * (float)Gd / ah + EPSF));
        float gcv = conf - 1.0f;
        contrib = gxv * gxv + gyv * gyv + gwv * gwv + ghv * ghv + gcv * gcv;
        for (int c = 0; c < CLS; ++c) {
            float pp = fast_sigmoid(x[base + (size_t)(5 + c) * cs]);
            float g = pp - ((c == label) ? 1.0f : 0.0f);
            contrib += g * g;
        }
    } else {
        // Non-target cell: only conf grad (zeroed if ignored by IOU)
        float gc = (maxiou > IGNORE_THRESH) ? 0.0f : conf;
        contrib = gc * gc;
    }

    // ---- wave32 reduction via V_WMMA_F32_16X16X4_F32 with ones-B ----
    // A[m][k]: lane m    -> {K0 = contrib, K1 = 0}
    //          lane m+16 -> {K2 = contrib, K3 = 0}
    // B = all ones  =>  D[m][n] = contrib(m) + contrib(m+16)
    // EXEC is all-1s here: full blocks, reconverged after the branch.
    v2f aop; aop.x = contrib; aop.y = 0.0f;
    v2f bop; bop.x = 1.0f;    bop.y = 1.0f;
    v8f c8 = {};
    c8 = __builtin_amdgcn_wmma_f32_16x16x4_f32(
        /*neg_a=*/false, aop, /*neg_b=*/false, bop,
        /*c_mod=*/(short)0, c8, /*reuse_a=*/false, /*reuse_b=*/false);
    // Lanes 0-15 hold D rows 0..7, lanes 16-31 hold rows 8..15 (column N = lane%16)
    float s8 = c8[0] + c8[1] + c8[2] + c8[3] + c8[4] + c8[5] + c8[6] + c8[7];
    float other = __shfl(s8, 16, 32);   // rows 8..15 partial from upper half
    if ((threadIdx.x & 31) == 0)
        atomicAdd(out, s8 + other);     // full wave sum
}

extern "C" void kernel_launch(void* const* d_in, const int* in_sizes, int n_in,
                              void* d_out, int out_size, void* d_ws, size_t ws_size,
                              hipStream_t stream) {
    const float* x       = (const float*)d_in[0];   // (16, 255, 52, 52)
    const float* anchors = (const float*)d_in[1];   // (3, 2)
    const float* target  = (const float*)d_in[2];   // (256, 7)
    float* out = (float*)d_out;                     // scalar
    int* map = (int*)d_ws;                          // NCELL ints (~507 KB)

    const int nblk = NCELL / 256;                   // 507, exact
    yolo_init_kernel<<<nblk, 256, 0, stream>>>(map, out);
    yolo_assign_kernel<<<1, NT, 0, stream>>>(target, map);
    yolo_grad_sq_kernel<<<nblk, 256, 0, stream>>>(x, anchors, target, map, out);
}